// GCN_estimator_63745904607632
// MI455X (gfx1250) — compile-verified
//
#include <hip/hip_runtime.h>
#include <math.h>

typedef float v2f __attribute__((ext_vector_type(2)));
typedef float v8f __attribute__((ext_vector_type(8)));

__device__ __forceinline__ v8f v8f_zero() {
    v8f z;
    #pragma unroll
    for (int i = 0; i < 8; i++) z[i] = 0.f;
    return z;
}

// D = A(16x4 f32) * B(4x16 f32) + C(16x16 f32)
__device__ __forceinline__ v8f wmma_f32(v2f a, v2f b, v8f c) {
    return __builtin_amdgcn_wmma_f32_16x16x4_f32(
        /*neg_a=*/false, a, /*neg_b=*/false, b,
        /*c_mod=*/(short)0, c, /*reuse_a=*/false, /*reuse_b=*/false);
}

// ---------------------------------------------------------------------------
// Kernel 1: sim chunk GEMM.  simChunk[r - rowOff][c] = F[r]·F[c]
// Block tile 128x128, 8 waves of 32x64, K staged in LDS (stride 20 pad).
// ---------------------------------------------------------------------------
#define SLDA 20
__global__ __launch_bounds__(256) void sim_gemm_kernel(
    const float* __restrict__ F, float* __restrict__ simChunk, int rowOff)
{
    __shared__ float At[128 * SLDA];
    __shared__ float Bt[128 * SLDA];
    const int tid  = threadIdx.x;
    const int wave = tid >> 5, lane = tid & 31;
    const int m0g  = rowOff + blockIdx.y * 128;       // global A row base
    const int n0g  = blockIdx.x * 128;                // global B row base (cols of sim)
    const int wm   = (wave >> 1) * 32;                // wave tile 32x64
    const int wn   = (wave & 1) * 64;
    const int lm   = lane & 15;
    const int lk   = (lane >> 4) * 2;

    v8f acc[2][4];
    #pragma unroll
    for (int i = 0; i < 2; i++)
        #pragma unroll
        for (int j = 0; j < 4; j++) acc[i][j] = v8f_zero();

    for (int kk = 0; kk < 2048; kk += 16) {
        __syncthreads();
        #pragma unroll
        for (int rep = 0; rep < 2; rep++) {
            int idx4 = tid + rep * 256;               // 0..511 float4 slots
            int r  = idx4 >> 2;
            int c4 = idx4 & 3;
            float4 va = *(const float4*)(F + (size_t)(m0g + r) * 2048 + kk + c4 * 4);
            float4 vb = *(const float4*)(F + (size_t)(n0g + r) * 2048 + kk + c4 * 4);
            *(float4*)&At[r * SLDA + c4 * 4] = va;
            *(float4*)&Bt[r * SLDA + c4 * 4] = vb;
        }
        __syncthreads();
        #pragma unroll
        for (int k4 = 0; k4 < 16; k4 += 4) {
            v2f a[2], b[4];
            #pragma unroll
            for (int i = 0; i < 2; i++)
                a[i] = *(const v2f*)&At[(wm + i * 16 + lm) * SLDA + k4 + lk];
            #pragma unroll
            for (int j = 0; j < 4; j++)
                b[j] = *(const v2f*)&Bt[(wn + j * 16 + lm) * SLDA + k4 + lk];
            #pragma unroll
            for (int i = 0; i < 2; i++)
                #pragma unroll
                for (int j = 0; j < 4; j++)
                    acc[i][j] = wmma_f32(a[i], b[j], acc[i][j]);
        }
    }
    const int r8 = (lane >> 4) * 8;
    #pragma unroll
    for (int i = 0; i < 2; i++)
        #pragma unroll
        for (int j = 0; j < 4; j++)
            #pragma unroll
            for (int r = 0; r < 8; r++) {
                int m = blockIdx.y * 128 + wm + i * 16 + r + r8;   // chunk-local row
                int n = n0g + wn + j * 16 + lm;
                simChunk[(size_t)m * 8192 + n] = acc[i][j][r];
            }
}

// ---------------------------------------------------------------------------
// Kernel 2: per-row top-64 (indices) + first-10 values.  One block per row.
// ---------------------------------------------------------------------------
__global__ __launch_bounds__(256) void topk_kernel(
    const float* __restrict__ simChunk, int rowOff,
    int* __restrict__ ori0, float* __restrict__ topv10)
{
    __shared__ float sv[8192];
    __shared__ float rv[8];
    __shared__ int   ri[8];
    const int lrow = blockIdx.x;
    const int row  = rowOff + lrow;
    const float* srow = simChunk + (size_t)lrow * 8192;
    for (int i = threadIdx.x; i < 8192; i += 256) sv[i] = srow[i];
    __syncthreads();

    for (int it = 0; it < 64; it++) {
        float bv = -INFINITY; int bi = 0x7fffffff;
        for (int i = threadIdx.x; i < 8192; i += 256) {
            float v = sv[i];
            if (v > bv || (v == bv && i < bi)) { bv = v; bi = i; }
        }
        #pragma unroll
        for (int off = 16; off > 0; off >>= 1) {
            float ov = __shfl_down(bv, off);
            int   oi = __shfl_down(bi, off);
            if (ov > bv || (ov == bv && oi < bi)) { bv = ov; bi = oi; }
        }
        if ((threadIdx.x & 31) == 0) { rv[threadIdx.x >> 5] = bv; ri[threadIdx.x >> 5] = bi; }
        __syncthreads();
        if (threadIdx.x == 0) {
            for (int w = 1; w < 8; w++)
                if (rv[w] > bv || (rv[w] == bv && ri[w] < bi)) { bv = rv[w]; bi = ri[w]; }
            ori0[(size_t)row * 64 + it] = bi;
            if (it < 10) topv10[(size_t)row * 10 + it] = bv;
            sv[bi] = -INFINITY;
        }
        __syncthreads();
    }
}

// ---------------------------------------------------------------------------
// Kernel 3: exp_sim[n] = exp( sum(top-10 values) / 40 )
// ---------------------------------------------------------------------------
__global__ void expsim_kernel(const float* __restrict__ topv10, float* __restrict__ exps)
{
    int n = blockIdx.x * 256 + threadIdx.x;
    if (n < 8192) {
        float s = 0.f;
        #pragma unroll
        for (int j = 0; j < 10; j++) s += topv10[(size_t)n * 10 + j];
        exps[n] = expf(s * (1.0f / 40.0f));
    }
}

// ---------------------------------------------------------------------------
// Kernel 4: build adjacency (sparse membership + softmax) and x = feat-feat[0]
// One block per graph.
// ---------------------------------------------------------------------------
__global__ __launch_bounds__(256) void build_graph_kernel(
    const int* __restrict__ indexes, const float* __restrict__ F,
    const int* __restrict__ ori0, const float* __restrict__ topv10,
    const float* __restrict__ exps, float* __restrict__ adjs, float* __restrict__ xs)
{
    __shared__ int   s_idx;
    __shared__ int   s_nbr[64];
    __shared__ int   s_neigh[64][10];
    __shared__ float s_tval[64][10];
    __shared__ float s_adj[64][64];
    const int b = blockIdx.x, tid = threadIdx.x;

    if (tid == 0) s_idx = indexes[b];
    __syncthreads();
    if (tid < 64) s_nbr[tid] = ori0[(size_t)s_idx * 64 + tid];
    __syncthreads();
    for (int e = tid; e < 640; e += 256) {
        int i = e / 10, j = e % 10;
        int q = ori0[(size_t)s_nbr[i] * 64 + j];
        s_neigh[i][j] = q;
        // A[i, neigh[i][j]] = topv(row nbr[i], rank j) * exp_sim[q]
        s_tval[i][j] = topv10[(size_t)s_nbr[i] * 10 + j] * exps[q];
    }
    __syncthreads();
    for (int e = tid; e < 4096; e += 256) {
        int i = e >> 6, l = e & 63;
        float s = 0.f;
        #pragma unroll
        for (int j = 0; j < 10; j++) {
            int q = s_neigh[i][j];
            bool found = false;
            #pragma unroll
            for (int j2 = 0; j2 < 10; j2++)
                if (s_neigh[l][j2] == q && s_tval[l][j2] > 0.f) found = true;
            if (found) s += s_tval[i][j];
        }
        s_adj[i][l] = s;
    }
    __syncthreads();
    if (tid < 64) {                                   // softmax over axis=1
        float mx = -INFINITY;
        for (int l = 0; l < 64; l++) mx = fmaxf(mx, s_adj[tid][l]);
        float sum = 0.f;
        for (int l = 0; l < 64; l++) { float e2 = expf(s_adj[tid][l] - mx); s_adj[tid][l] = e2; sum += e2; }
        float inv = 1.f / sum;
        for (int l = 0; l < 64; l++) s_adj[tid][l] *= inv;
    }
    __syncthreads();
    for (int e = tid; e < 4096; e += 256) adjs[(size_t)b * 4096 + e] = (&s_adj[0][0])[e];

    const float* f0 = F + (size_t)s_nbr[0] * 2048;
    for (int i = 0; i < 64; i++) {
        const float* fi = F + (size_t)s_nbr[i] * 2048;
        float* xo = xs + ((size_t)b * 64 + i) * 2048;
        for (int d = tid * 4; d < 2048; d += 1024) {
            float4 a = *(const float4*)(fi + d);
            float4 c = *(const float4*)(f0 + d);
            float4 r; r.x = a.x - c.x; r.y = a.y - c.y; r.z = a.z - c.z; r.w = a.w - c.w;
            *(float4*)(xo + d) = r;
        }
    }
}

// ---------------------------------------------------------------------------
// Kernel 5: batched AGG[b] = adj[b](64x64) @ X[b](64xD).  Block: 64 x 128 tile.
// ---------------------------------------------------------------------------
#define ALDA 68
__global__ __launch_bounds__(256) void bmm_adj_kernel(
    const float* __restrict__ adjs, const float* __restrict__ X,
    float* __restrict__ AGG, int D)
{
    __shared__ float adjL[64 * ALDA];
    const int b  = blockIdx.y;
    const int n0 = blockIdx.x * 128;
    const float* adjp = adjs + (size_t)b * 4096;
    const float* Xp   = X   + (size_t)b * 64 * D;
    float*       Op   = AGG + (size_t)b * 64 * D;
    const int tid = threadIdx.x, wave = tid >> 5, lane = tid & 31;
    const int lm = lane & 15, lk = (lane >> 4) * 2;
    const int m0w = (wave >> 2) * 32;
    const int n0w = n0 + (wave & 3) * 32;

    for (int e = tid; e < 1024; e += 256) {
        int m = e >> 4, c4 = e & 15;
        float4 v = *(const float4*)(adjp + m * 64 + c4 * 4);
        *(float4*)&adjL[m * ALDA + c4 * 4] = v;
    }
    __syncthreads();

    v8f acc[2][2];
    #pragma unroll
    for (int i = 0; i < 2; i++)
        #pragma unroll
        for (int j = 0; j < 2; j++) acc[i][j] = v8f_zero();

    for (int k = 0; k < 64; k += 4) {
        v2f a[2], bb[2];
        #pragma unroll
        for (int i = 0; i < 2; i++)
            a[i] = *(const v2f*)&adjL[(m0w + i * 16 + lm) * ALDA + k + lk];
        #pragma unroll
        for (int j = 0; j < 2; j++) {
            int n = n0w + j * 16 + lm;
            int kr = k + lk;
            v2f t; t.x = Xp[(size_t)kr * D + n]; t.y = Xp[(size_t)(kr + 1) * D + n];
            bb[j] = t;
        }
        #pragma unroll
        for (int i = 0; i < 2; i++)
            #pragma unroll
            for (int j = 0; j < 2; j++)
                acc[i][j] = wmma_f32(a[i], bb[j], acc[i][j]);
    }
    const int r8 = (lane >> 4) * 8;
    #pragma unroll
    for (int i = 0; i < 2; i++)
        #pragma unroll
        for (int j = 0; j < 2; j++)
            #pragma unroll
            for (int r = 0; r < 8; r++) {
                int m = m0w + i * 16 + r + r8;
                int n = n0w + j * 16 + lm;
                Op[(size_t)m * D + n] = acc[i][j][r];
            }
}

// ---------------------------------------------------------------------------
// Kernel 6: OUT[b](64xN) = act( A1[b](64xK)@W[0:K] + A2[b](64xK)@W[K:2K] + bias )
// act: 0=none, 1=relu, 2=prelu(pa).  Tile 64x64.
// ---------------------------------------------------------------------------
#define GLDA 20
__global__ __launch_bounds__(256) void gemm2_kernel(
    const float* __restrict__ A1, const float* __restrict__ A2,
    const float* __restrict__ W, const float* __restrict__ bias,
    const float* __restrict__ pa, float* __restrict__ OUT,
    int K, int N, int act)
{
    __shared__ float At[64 * GLDA];
    __shared__ float Wt[64 * GLDA];     // transposed: [n-within-64][k-within-16]
    const int b  = blockIdx.y;
    const int n0 = blockIdx.x * 64;
    const int tid = threadIdx.x, wave = tid >> 5, lane = tid & 31;
    const int lm = lane & 15, lk = (lane >> 4) * 2;
    const int mi  = (wave >> 1) * 16;
    const int nj0 = (wave & 1) * 32;

    v8f acc[2];
    acc[0] = v8f_zero(); acc[1] = v8f_zero();

    const int npass = A2 ? 2 : 1;
    for (int pass = 0; pass < npass; pass++) {
        const float* A  = (pass ? A2 : A1) + (size_t)b * 64 * K;
        const float* Wp = W + (size_t)pass * K * N;
        for (int kk = 0; kk < K; kk += 16) {
            __syncthreads();
            {   // stage A: 64 rows x 16 cols
                int m = tid >> 2, c4 = tid & 3;
                float4 v = *(const float4*)(A + (size_t)m * K + kk + c4 * 4);
                *(float4*)&At[m * GLDA + c4 * 4] = v;
            }
            {   // stage W transposed
                int kr = tid >> 4, c4 = tid & 15;
                float4 v = *(const float4*)(Wp + (size_t)(kk + kr) * N + n0 + c4 * 4);
                Wt[(c4 * 4 + 0) * GLDA + kr] = v.x;
                Wt[(c4 * 4 + 1) * GLDA + kr] = v.y;
                Wt[(c4 * 4 + 2) * GLDA + kr] = v.z;
                Wt[(c4 * 4 + 3) * GLDA + kr] = v.w;
            }
            __syncthreads();
            #pragma unroll
            for (int k4 = 0; k4 < 16; k4 += 4) {
                v2f a  = *(const v2f*)&At[(mi + lm) * GLDA + k4 + lk];
                v2f b0 = *(const v2f*)&Wt[(nj0 + lm) * GLDA + k4 + lk];
                v2f b1 = *(const v2f*)&Wt[(nj0 + 16 + lm) * GLDA + k4 + lk];
                acc[0] = wmma_f32(a, b0, acc[0]);
                acc[1] = wmma_f32(a, b1, acc[1]);
            }
        }
    }
    const int r8 = (lane >> 4) * 8;
    #pragma unroll
    for (int j = 0; j < 2; j++) {
        int n = n0 + nj0 + j * 16 + lm;
        float bv = bias[n];
        float av = (act == 2) ? pa[n] : 0.f;
        #pragma unroll
        for (int r = 0; r < 8; r++) {
            int m = mi + r + r8;
            float h = acc[j][r] + bv;
            if (act == 1)      h = fmaxf(h, 0.f);
            else if (act == 2) h = (h > 0.f) ? h : av * h;
            OUT[((size_t)b * 64 + m) * N + n] = h;
        }
    }
}

// ---------------------------------------------------------------------------
// Kernel 7: pred = h5 @ Wc2(256x2) + bc2
// ---------------------------------------------------------------------------
__global__ __launch_bounds__(128) void head2_kernel(
    const float* __restrict__ h5, const float* __restrict__ Wc2,
    const float* __restrict__ bc2, float* __restrict__ out)
{
    const int b = blockIdx.x, t = threadIdx.x;
    const int i = t >> 1, c = t & 1;
    const float* hp = h5 + ((size_t)b * 64 + i) * 256;
    float s = bc2[c];
    for (int d = 0; d < 256; d++) s += hp[d] * Wc2[d * 2 + c];
    out[((size_t)b * 64 + i) * 2 + c] = s;
}

// ---------------------------------------------------------------------------
extern "C" void kernel_launch(void* const* d_in, const int* in_sizes, int n_in,
                              void* d_out, int out_size, void* d_ws, size_t ws_size,
                              hipStream_t stream)
{
    (void)in_sizes; (void)n_in; (void)out_size; (void)ws_size;
    const int*   indexes = (const int*)  d_in[0];
    const float* F       = (const float*)d_in[1];
    const float* W1  = (const float*)d_in[4];
    const float* b1  = (const float*)d_in[5];
    const float* W2  = (const float*)d_in[6];
    const float* b2  = (const float*)d_in[7];
    const float* W3  = (const float*)d_in[8];
    const float* b3  = (const float*)d_in[9];
    const float* W4  = (const float*)d_in[10];
    const float* b4  = (const float*)d_in[11];
    const float* Wc1 = (const float*)d_in[12];
    const float* bc1 = (const float*)d_in[13];
    const float* pa  = (const float*)d_in[14];
    const float* Wc2 = (const float*)d_in[15];
    const float* bc2 = (const float*)d_in[16];

    char* ws = (char*)d_ws;
    const size_t MB = 1024ull * 1024ull;
    int*   ori0 = (int*)  (ws + 0);          //  2 MB
    float* topv = (float*)(ws + 2 * MB);     //  320 KB
    float* exps = (float*)(ws + 3 * MB);     //  32 KB
    float* adjs = (float*)(ws + 4 * MB);     //  1 MB
    float* xs   = (float*)(ws + 5 * MB);     //  32 MB
    float* simc = (float*)(ws + 37 * MB);    //  16 MB (reused as agg)
    float* agg  = (float*)(ws + 37 * MB);    //  32 MB
    float* h1   = (float*)(ws + 69 * MB);    //  16 MB
    float* h2   = (float*)(ws + 85 * MB);    //  8 MB
    float* h3   = (float*)(ws + 93 * MB);    //  4 MB
    float* h4   = (float*)(ws + 97 * MB);    //  4 MB
    float* h5   = (float*)(ws + 101 * MB);   //  4 MB -> 105 MB total

    const int CHUNK = 512;
    for (int rowOff = 0; rowOff < 8192; rowOff += CHUNK) {
        sim_gemm_kernel<<<dim3(8192 / 128, CHUNK / 128), 256, 0, stream>>>(F, simc, rowOff);
        topk_kernel<<<CHUNK, 256, 0, stream>>>(simc, rowOff, ori0, topv);
    }
    expsim_kernel<<<32, 256, 0, stream>>>(topv, exps);
    build_graph_kernel<<<64, 256, 0, stream>>>(indexes, F, ori0, topv, exps, adjs, xs);

    // layer 1: D=2048 -> 1024
    bmm_adj_kernel<<<dim3(2048 / 128, 64), 256, 0, stream>>>(adjs, xs, agg, 2048);
    gemm2_kernel<<<dim3(1024 / 64, 64), 256, 0, stream>>>(xs, agg, W1, b1, nullptr, h1, 2048, 1024, 1);
    // layer 2: 1024 -> 512
    bmm_adj_kernel<<<dim3(1024 / 128, 64), 256, 0, stream>>>(adjs, h1, agg, 1024);
    gemm2_kernel<<<dim3(512 / 64, 64), 256, 0, stream>>>(h1, agg, W2, b2, nullptr, h2, 1024, 512, 1);
    // layer 3: 512 -> 256
    bmm_adj_kernel<<<dim3(512 / 128, 64), 256, 0, stream>>>(adjs, h2, agg, 512);
    gemm2_kernel<<<dim3(256 / 64, 64), 256, 0, stream>>>(h2, agg, W3, b3, nullptr, h3, 512, 256, 1);
    // layer 4: 256 -> 256
    bmm_adj_kernel<<<dim3(256 / 128, 64), 256, 0, stream>>>(adjs, h3, agg, 256);
    gemm2_kernel<<<dim3(256 / 64, 64), 256, 0, stream>>>(h3, agg, W4, b4, nullptr, h4, 256, 256, 1);
    // head: Wc1 + prelu, then Wc2
    gemm2_kernel<<<dim3(256 / 64, 64), 256, 0, stream>>>(h4, nullptr, Wc1, bc1, pa, h5, 256, 256, 2);
    head2_kernel<<<64, 128, 0, stream>>>(h5, Wc2, bc2, (float*)d_out);
}